// ProposalLayer_48627619725486
// MI455X (gfx1250) — compile-verified
//
#include <hip/hip_runtime.h>
#include <stdint.h>

typedef unsigned int u32;
typedef unsigned long long u64;

#define B_        8
#define N_        262144
#define PRE_      6000
#define PROP_     1000
#define NMS_THR_  0.7f
#define TOPK_STRIDE 6016          // padded so box tile reads stay in-bounds
#define NWORDS    188             // ceil(6000/32); last word has 16 valid bits
#define SORT_N    8192

// ---------------------------------------------------------------------------
// Kernel 1 (fused): per-batch exact top-6000 radix select (3 levels of
// 11/11/10 bits on the float bit pattern; scores in [0,1) so uint order ==
// float order), stable compaction directly into the LDS sort array (ties at
// the threshold take lowest indices, matching jax.lax.top_k), bitonic sort of
// 8192 packed (score<<32 | ~index) keys, then gather + box-delta decode.
// LDS budget: exactly 64 KB; histogram and scan scratch alias dead regions of
// the sort array.
// ---------------------------------------------------------------------------
__global__ __launch_bounds__(1024) void k_topk(const float* __restrict__ probs,
                                               const float* __restrict__ bbox,
                                               const float* __restrict__ anchors,
                                               float4* __restrict__ boxes) {
  const int b   = blockIdx.x;
  const int tid = threadIdx.x;
  __shared__ u64 key[SORT_N];                 // 65536 B, multi-purpose
  u32* hist    = (u32*)key;                   // u32[0..2047]  (select phase only)
  u32* scratch = (u32*)key + 12032;           // u32[12032..]  (= u64 idx 6016+, dead
                                              //  until the pre-sort pad overwrites it)
  // scratch map: [0..63] chunk sums | [64] digit | [65] above | [66] totG
  //              [67] totE | [68..99] waveG | [100..131] waveE
  //              [132..163] waveGo | [164..195] waveEo

  const float2* sp2 = (const float2*)(probs + (size_t)b * N_ * 2);  // score = .y

  // ---- 3-level radix select for the exact 6000th-largest key ----
  u32 prefix = 0, maskhi = 0, krem = PRE_, cntgt = 0;
#pragma unroll
  for (int level = 0; level < 3; ++level) {
    const int shift = (level == 0) ? 21 : (level == 1) ? 10 : 0;
    const int nb    = (level == 2) ? 1024 : 2048;
    for (int t = tid; t < nb; t += 1024) hist[t] = 0;
    __syncthreads();
    for (int i = tid; i < N_; i += 1024) {
      __builtin_prefetch(&sp2[i + 4096], 0, 1);        // global_prefetch_b8
      const u32 k = __float_as_uint(sp2[i].y);
      if ((k & maskhi) == prefix)
        atomicAdd(&hist[(k >> shift) & (u32)(nb - 1)], 1u);
    }
    __syncthreads();
    const int per = nb >> 6;                           // 64 chunks
    if (tid < 64) {
      u32 s = 0;
      for (int q = 0; q < per; ++q) s += hist[tid * per + q];
      scratch[tid] = s;
    }
    __syncthreads();
    if (tid == 0) {
      u32 cum = 0; int c = 63;
      for (; c >= 0; --c) { const u32 v = scratch[c]; if (cum + v >= krem) break; cum += v; }
      if (c < 0) c = 0;
      int d = (c + 1) * per - 1;
      for (; d >= c * per; --d) { const u32 h = hist[d]; if (cum + h >= krem) break; cum += h; }
      if (d < c * per) d = c * per;
      scratch[64] = (u32)d;
      scratch[65] = cum;
    }
    __syncthreads();
    const u32 digit = scratch[64];
    const u32 above = scratch[65];
    prefix |= digit << shift;
    maskhi |= (u32)(nb - 1) << shift;
    krem  -= above;
    cntgt += above;
    __syncthreads();
  }
  const u32 T       = prefix;   // exact 6000th-largest key
  const u32 need_eq = krem;     // ties to take (>=1), smallest indices first
  const u32 base_eq = cntgt;    // #keys strictly greater than T

  // ---- stable compaction straight into the LDS sort array ----
  u32* waveG  = scratch + 68;
  u32* waveE  = scratch + 100;
  u32* waveGo = scratch + 132;
  u32* waveEo = scratch + 164;
  const int lane = tid & 31, wave = tid >> 5;
  const u32 lmask = (1u << lane) - 1u;
  u32 runG = 0, runE = 0;
  for (int base = 0; base < N_; base += 1024) {
    const int i = base + tid;
    const u32 k = __float_as_uint(sp2[i].y);
    const bool pg = k > T;
    const bool pe = (k == T);
    const u32 mg = (u32)__ballot(pg);
    const u32 me = (u32)__ballot(pe);
    if (lane == 0) { waveG[wave] = __popc(mg); waveE[wave] = __popc(me); }
    __syncthreads();
    if (tid == 0) {
      u32 sg = 0, se = 0;
      for (int w = 0; w < 32; ++w) {
        const u32 g = waveG[w], e = waveE[w];
        waveGo[w] = sg; waveEo[w] = se; sg += g; se += e;
      }
      scratch[66] = sg; scratch[67] = se;
    }
    __syncthreads();
    if (pg)
      key[runG + waveGo[wave] + __popc(mg & lmask)] = ((u64)k << 32) | (~(u32)i);
    if (pe) {
      const u32 r = runE + waveEo[wave] + __popc(me & lmask);
      if (r < need_eq) key[base_eq + r] = ((u64)k << 32) | (~(u32)i);
    }
    runG += scratch[66]; runE += scratch[67];
    __syncthreads();
  }

  // ---- pad + bitonic sort (descending: score desc, index asc on ties) ----
  for (int r = PRE_ + tid; r < SORT_N; r += 1024) key[r] = 0ull;
  __syncthreads();
  for (int k = 2; k <= SORT_N; k <<= 1) {
    for (int j = k >> 1; j > 0; j >>= 1) {
      for (int t = tid; t < SORT_N; t += 1024) {
        const int ixj = t ^ j;
        if (ixj > t) {
          const u64 a = key[t], c = key[ixj];
          const bool up = ((t & k) == 0);
          if (up ? (a < c) : (a > c)) { key[t] = c; key[ixj] = a; }
        }
      }
      __syncthreads();
    }
  }

  // ---- gather + decode + clip ----
  const float4* anc4 = (const float4*)anchors + (size_t)b * N_;
  const float4* dl4  = (const float4*)bbox    + (size_t)b * N_;
  float4* ob = boxes + (size_t)b * TOPK_STRIDE;
  for (int r = tid; r < PRE_; r += 1024) {
    const u64 kv  = key[r];
    const u32 idx = ~(u32)(kv & 0xFFFFFFFFull);
    const float4 a = anc4[idx];
    const float4 d = dl4[idx];
    const float d0 = d.x * 0.1f, d1 = d.y * 0.1f;
    const float d2 = d.z * 0.2f, d3 = d.w * 0.2f;
    float h = a.z - a.x, w = a.w - a.y;
    float cy = a.x + 0.5f * h, cx = a.y + 0.5f * w;
    cy += d0 * h; cx += d1 * w;
    h *= expf(d2); w *= expf(d3);
    const float y1 = fminf(fmaxf(cy - 0.5f * h, 0.f), 1.f);
    const float x1 = fminf(fmaxf(cx - 0.5f * w, 0.f), 1.f);
    const float y2 = fminf(fmaxf(cy + 0.5f * h, 0.f), 1.f);
    const float x2 = fminf(fmaxf(cx + 0.5f * w, 0.f), 1.f);
    ob[r] = make_float4(y1, x1, y2, x2);
  }
}

// ---------------------------------------------------------------------------
// Kernel 2: pairwise suppression bitmask. Block = 256 rows (i) x one 32-column
// word (w). The 32 column boxes (512 B) are staged into LDS via the CDNA5
// async-tensor path (global_load_async_to_lds_b128 + s_wait_asynccnt).
// ---------------------------------------------------------------------------
__global__ __launch_bounds__(256) void k_mask(const float4* __restrict__ boxes,
                                              u32* __restrict__ mask) {
  const int w = blockIdx.x;                     // 0..187
  const int b = blockIdx.z;
  const int i = blockIdx.y * 256 + threadIdx.x;
  __shared__ float4 jb[32];
  const float4* bb = boxes + (size_t)b * TOPK_STRIDE;
  const int jbase = w * 32;
  if (threadIdx.x < 32) {
    const float4* src = bb + jbase + threadIdx.x;   // padded stride: in-bounds
    const u32 lds = (u32)(uintptr_t)(&jb[threadIdx.x]);
    const u64 ga  = (u64)(uintptr_t)src;
    asm volatile("global_load_async_to_lds_b128 %0, %1, off"
                 :: "v"(lds), "v"(ga) : "memory");
    asm volatile("s_wait_asynccnt 0" ::: "memory");
  }
  __syncthreads();
  if (i >= PRE_) return;
  const float4 bi = bb[i];
  const float area_i = (bi.z - bi.x) * (bi.w - bi.y);
  u32 bits = 0;
#pragma unroll
  for (int t = 0; t < 32; ++t) {
    const int j = jbase + t;
    const float4 bj = jb[t];
    const float iy1 = fmaxf(bi.x, bj.x), ix1 = fmaxf(bi.y, bj.y);
    const float iy2 = fminf(bi.z, bj.z), ix2 = fminf(bi.w, bj.w);
    const float inter = fmaxf(iy2 - iy1, 0.f) * fmaxf(ix2 - ix1, 0.f);
    const float area_j = (bj.z - bj.x) * (bj.w - bj.y);
    const float uni = fmaxf(area_i + area_j - inter, 1e-10f);
    const bool sup = inter > NMS_THR_ * uni;    // iou > 0.7
    if (j < PRE_ && j != i && sup) bits |= (1u << t);
  }
  mask[((size_t)b * PRE_ + i) * NWORDS + w] = bits;
}

// ---------------------------------------------------------------------------
// Kernel 3: serial greedy resolve, one wave32 per batch (single-wave workgroup
// => barriers are free). Word-wise skip over suppressed candidates via
// __ffs(~removed) so serial steps ~ kept + NWORDS instead of 6000. The kept
// bit is folded into the lane-parallel suppression OR (no intra-wave RMW race).
// ---------------------------------------------------------------------------
__global__ __launch_bounds__(32) void k_nms(const float4* __restrict__ boxes,
                                            const u32* __restrict__ mask,
                                            float* __restrict__ out) {
  const int b = blockIdx.x;
  const int lane = threadIdx.x;
  __shared__ u32 removed[NWORDS];
  for (int w = lane; w < NWORDS; w += 32) removed[w] = 0u;
  __syncthreads();
  const float4* bb   = boxes + (size_t)b * TOPK_STRIDE;
  const u32*    mrow = mask  + (size_t)b * PRE_ * NWORDS;
  float4* ob = (float4*)(out + (size_t)b * PROP_ * 4);
  int kept = 0;
  for (int wd = 0; wd < NWORDS && kept < PROP_; ++wd) {
    const u32 valid = (wd == NWORDS - 1) ? 0xFFFFu : 0xFFFFFFFFu;  // 6000=187*32+16
    for (;;) {
      const u32 avail = (~removed[wd]) & valid;      // uniform broadcast read
      if (avail == 0u) break;
      const int t = __ffs(avail) - 1;                // next live candidate
      const int i = wd * 32 + t;
      if (lane == 0) ob[kept] = bb[i];
      const u32* row = mrow + (size_t)i * NWORDS;
      for (int w = lane; w < NWORDS; w += 32) {
        u32 v = row[w];
        if (w == wd) v |= (1u << t);                 // retire the kept candidate
        removed[w] |= v;
      }
      __syncthreads();
      if (++kept == PROP_) break;
    }
  }
  float* of = out + (size_t)b * PROP_ * 4;
  for (int idx = kept * 4 + lane; idx < PROP_ * 4; idx += 32) of[idx] = 0.f;
}

// ---------------------------------------------------------------------------
extern "C" void kernel_launch(void* const* d_in, const int* in_sizes, int n_in,
                              void* d_out, int out_size, void* d_ws, size_t ws_size,
                              hipStream_t stream) {
  (void)in_sizes; (void)n_in; (void)out_size; (void)ws_size;
  const float* probs = (const float*)d_in[0];   // (B,N,2)
  const float* bbox  = (const float*)d_in[1];   // (B,N,4)
  const float* anch  = (const float*)d_in[2];   // (B,N,4)
  float* out = (float*)d_out;                   // (B,1000,4)

  char* ws = (char*)d_ws;
  float4* boxes = (float4*)ws;                                   // B*6016 float4
  size_t  maskOff = (size_t)B_ * TOPK_STRIDE * sizeof(float4);
  u32*    mask  = (u32*)(ws + maskOff);                          // B*6000*188 u32 (~36 MB)

  hipLaunchKernelGGL(k_topk, dim3(B_), dim3(1024), 0, stream, probs, bbox, anch, boxes);
  hipLaunchKernelGGL(k_mask, dim3(NWORDS, (PRE_ + 255) / 256, B_), dim3(256), 0, stream, boxes, mask);
  hipLaunchKernelGGL(k_nms,  dim3(B_), dim3(32), 0, stream, boxes, mask, out);
}